// ECGKANModel_28733331210514
// MI455X (gfx1250) — compile-verified
//
#include <hip/hip_runtime.h>
#include <math.h>

#define SEQ 256
#define NB  8          // batch
#define NCOEF 67

typedef __attribute__((ext_vector_type(2))) float v2f;
typedef __attribute__((ext_vector_type(8))) float v8f;

static __device__ __forceinline__ float dsilu(float x) { return x / (1.0f + __expf(-x)); }

// ---------------------------------------------------------------------------
// Batched WMMA fp32 GEMM (NT form):  C[M,N] = A[M,K] * Bt[N,K]^T  (+ bias)
//   A  : per-batch, row-major [M,K], stride sAb
//   Bt : row-major [N,K] (i.e. B already transposed), selected by
//        (bmod ? batch%bmod : batch)*sBb
//   bias modes: 0 none; 1 bias[(batch%bmod)*M*N + r*N + n]; 2 bias[(batch%bmod)*N + n]
// One wave per 16x16 C tile. K stepped by 16 with 4 independent accumulators
// (every K used here is a multiple of 16). All fragment loads are contiguous
// 8-byte v2f loads. FULLN==1 -> N%16==0 (no edge masking, no branches).
// ---------------------------------------------------------------------------
template<int FULLN>
__global__ __launch_bounds__(32)
void wmma_gemm_nt(const float* __restrict__ A, const float* __restrict__ Bt,
                  const float* __restrict__ bias, float* __restrict__ C,
                  int M, int N, int K,
                  long sAb, long sBb, long sCb,
                  int bmod, int biasMode)
{
    int batch = blockIdx.z;
    int row   = blockIdx.x * 16;
    int col   = blockIdx.y * 16;
    int lane  = threadIdx.x;            // 0..31
    int lo = lane & 15, hi = lane >> 4; // hi: 0 or 1

    const float* Ab = A + (long)batch * sAb;
    int bsel = bmod ? (batch % bmod) : batch;
    const float* Bb = Bt + (long)bsel * sBb;
    float*       Cb = C + (long)batch * sCb;

    int   n    = col + lo;
    bool  nok  = FULLN ? true : (n < N);
    int   nc   = FULLN ? n : (nok ? n : 0);       // clamped safe row of Bt
    float nmul = FULLN ? 1.0f : (nok ? 1.0f : 0.0f);

    // Fragment pointers: lanes 0-15 cover K 2*hi..2*hi+1 = {0,1}, lanes 16-31 {2,3}
    const float* ap = Ab + (long)(row + lo) * K + 2 * hi;
    const float* bp = Bb + (long)nc * K + 2 * hi;

    v8f acc0 = {0.f,0.f,0.f,0.f,0.f,0.f,0.f,0.f};
    v8f acc1 = acc0, acc2 = acc0, acc3 = acc0;

    for (int k0 = 0; k0 < K; k0 += 16) {
        v2f a0 = *(const v2f*)(ap);
        v2f a1 = *(const v2f*)(ap + 4);
        v2f a2 = *(const v2f*)(ap + 8);
        v2f a3 = *(const v2f*)(ap + 12);
        v2f b0 = *(const v2f*)(bp);
        v2f b1 = *(const v2f*)(bp + 4);
        v2f b2 = *(const v2f*)(bp + 8);
        v2f b3 = *(const v2f*)(bp + 12);
        if (!FULLN) {
            b0.x *= nmul; b0.y *= nmul; b1.x *= nmul; b1.y *= nmul;
            b2.x *= nmul; b2.y *= nmul; b3.x *= nmul; b3.y *= nmul;
        }
        acc0 = __builtin_amdgcn_wmma_f32_16x16x4_f32(false, a0, false, b0, (short)0, acc0, false, false);
        acc1 = __builtin_amdgcn_wmma_f32_16x16x4_f32(false, a1, false, b1, (short)0, acc1, false, false);
        acc2 = __builtin_amdgcn_wmma_f32_16x16x4_f32(false, a2, false, b2, (short)0, acc2, false, false);
        acc3 = __builtin_amdgcn_wmma_f32_16x16x4_f32(false, a3, false, b3, (short)0, acc3, false, false);
        ap += 16;
        bp += 16;
    }

    if (!nok) return;
#pragma unroll
    for (int j = 0; j < 8; ++j) {
        int r = row + j + 8 * hi;       // C layout: VGPR j -> M=j (lanes<16), M=j+8 (lanes>=16)
        float v = (acc0[j] + acc1[j]) + (acc2[j] + acc3[j]);
        if (biasMode == 1)      v += bias[((long)(batch % bmod) * M + r) * N + n];
        else if (biasMode == 2) v += bias[(long)(batch % bmod) * N + n];
        Cb[(long)r * N + n] = v;
    }
}

// Bt[batch][n][k] = B[batch][k][n]; grid (batches, K), block 256 over n
__global__ void transpose_kernel(const float* __restrict__ src0, float* __restrict__ dst0,
                                 int K, int N, long sB)
{
    int batch = blockIdx.x;
    int k = blockIdx.y;
    const float* src = src0 + (long)batch * sB + (long)k * N;
    float* dst = dst0 + (long)batch * sB + k;
    for (int n = threadIdx.x; n < N; n += blockDim.x)
        dst[(long)n * K] = src[n];
}

// ---------------------------------------------------------------------------
// Prep stage: conv2d + BN (+ optional SiLU) over 32x32 images, 5 or 32 in-ch.
// grid (B, Cout), block 256 (each thread 4 pixels)
// ---------------------------------------------------------------------------
__global__ void conv_bn_act(const float* __restrict__ x, const float* __restrict__ w,
                            const float* __restrict__ cb,
                            const float* __restrict__ g, const float* __restrict__ bb,
                            const float* __restrict__ m, const float* __restrict__ v,
                            float* __restrict__ out,
                            int Cin, int k, int pad, int coff, int outCtot, int doSilu)
{
    int b = blockIdx.x, oc = blockIdx.y, tid = threadIdx.x;
    float scale = g[oc] * rsqrtf(v[oc] + 1e-5f);
    float shift = bb[oc] - m[oc] * scale;
    for (int rep = 0; rep < 4; ++rep) {
        int p = tid + rep * 256;
        int h = p >> 5, wq = p & 31;
        float acc = cb[oc];
        for (int ic = 0; ic < Cin; ++ic) {
            const float* xi = x + ((long)b * Cin + ic) * 1024;
            const float* wk = w + ((long)oc * Cin + ic) * k * k;
            for (int dy = 0; dy < k; ++dy) {
                int hy = h + dy - pad;
                if (hy < 0 || hy > 31) continue;
                for (int dx = 0; dx < k; ++dx) {
                    int wx = wq + dx - pad;
                    if (wx < 0 || wx > 31) continue;
                    acc += wk[dy * k + dx] * xi[hy * 32 + wx];
                }
            }
        }
        float y = acc * scale + shift;
        if (doSilu) y = dsilu(y);
        out[(((long)b * outCtot + coff + oc) * 32 + h) * 32 + wq] = y;
    }
}

// silu(pc+sc) + PixelUnshuffle(2) + transpose -> x0[b][s][f], f=c*4+2*(h&1)+(w&1)
__global__ void prep_finish(const float* __restrict__ pc, const float* __restrict__ sc,
                            float* __restrict__ x0)
{
    int b = blockIdx.x, c = blockIdx.y, tid = threadIdx.x;
    for (int rep = 0; rep < 4; ++rep) {
        int p = tid + rep * 256;
        int h = p >> 5, wq = p & 31;
        long src = (((long)b * 32 + c) * 32 + h) * 32 + wq;
        float y = dsilu(pc[src] + sc[src]);
        int f = c * 4 + (h & 1) * 2 + (wq & 1);
        int s = (h >> 1) * 16 + (wq >> 1);
        x0[((long)b * SEQ + s) * 128 + f] = y;
    }
}

// ---------------------------------------------------------------------------
// KAN: per-edge statistics:  xn = clip(mean_f(x)/max_s|.|), a = mean_f|x|
// grid (B, in), block 256 (s)
// ---------------------------------------------------------------------------
__global__ void kan_stats(const float* __restrict__ x, float* __restrict__ xn,
                          float* __restrict__ amean, int in_n, int F)
{
    int b = blockIdx.x, i = blockIdx.y, s = threadIdx.x;
    const float* row = x + (((long)b * in_n + i) * SEQ + s) * F;
    float sum = 0.f, asum = 0.f;
    for (int f = 0; f < F; ++f) { float t = row[f]; sum += t; asum += fabsf(t); }
    float xe = sum / (float)F;
    __shared__ float sm[256];
    sm[s] = fabsf(xe);
    __syncthreads();
    for (int st = 128; st; st >>= 1) { if (s < st) sm[s] = fmaxf(sm[s], sm[s + st]); __syncthreads(); }
    float mx = sm[0];
    float v = xe / (mx + 1e-8f);
    v = fminf(0.99f, fmaxf(-0.99f, v));
    long o = ((long)b * in_n + i) * SEQ + s;
    xn[o] = v;
    amean[o] = asum / (float)F;
}

// edge[b,i,o,s] = silu(xn)*base_scales[i,o] + sum_c basis(xn,c)*sw[i,o,c]
// grid (B, in, out), block 256 (s)
__global__ void kan_edge(const float* __restrict__ xn, const float* __restrict__ sw,
                         const float* __restrict__ bs, float* __restrict__ E,
                         int in_n, int out_n)
{
    int b = blockIdx.x, i = blockIdx.y, o = blockIdx.z, s = threadIdx.x;
    float xv = xn[((long)b * in_n + i) * SEQ + s];
    const float hstep = 2.0f / 66.0f;
    const float* w = sw + ((long)i * out_n + o) * NCOEF;
    float acc = 0.f;
    for (int c = 0; c < NCOEF; ++c) {
        float gc = -1.0f + hstep * (float)c;
        float d = fabsf(xv - gc) / hstep;
        float bas;
        if (d < 1.0f)      bas = 2.0f / 3.0f - d * d + d * d * d * 0.5f;
        else if (d < 2.0f) { float u = 2.0f - d; bas = u * u * u * (1.0f / 6.0f); }
        else               bas = 0.0f;
        acc += bas * w[c];
    }
    E[(((long)b * in_n + i) * out_n + o) * SEQ + s] = dsilu(xv) * bs[(long)i * out_n + o] + acc;
}

// ee = mean_s(|E + |omiga|| * a); mask = sigmoid((ee-|tau|)/t); g = ee/(|tau|+1e-8)*mask
// grid (B, in, out), block 256 (s)
__global__ void kan_ee(const float* __restrict__ E, const float* __restrict__ a,
                       const float* __restrict__ omiga, const float* __restrict__ tau,
                       const float* __restrict__ temp, float* __restrict__ mask,
                       float* __restrict__ gbuf, int in_n, int out_n)
{
    int b = blockIdx.x, i = blockIdx.y, o = blockIdx.z, s = threadIdx.x;
    float wv = fabsf(omiga[(long)i * out_n + o]);
    float val = fabsf(E[(((long)b * in_n + i) * out_n + o) * SEQ + s] + wv)
              * a[((long)b * in_n + i) * SEQ + s];
    __shared__ float sm[256];
    sm[s] = val;
    __syncthreads();
    for (int st = 128; st; st >>= 1) { if (s < st) sm[s] += sm[s + st]; __syncthreads(); }
    if (s == 0) {
        float ee = sm[0] / (float)SEQ;
        float t  = fabsf(temp[0]) + 1e-4f;
        float tv = fabsf(tau[(long)i * out_n + o]);
        float mk = 1.0f / (1.0f + __expf(-(ee - tv) / t));
        long oi = ((long)b * in_n + i) * out_n + o;
        mask[oi] = mk;
        gbuf[oi] = ee / (tv + 1e-8f) * mk;
    }
}

// combined[b,o,s,f] = (1/in) * sum_i (E+|omiga|)*mask*x   ; 1 block per (b,o,s), F threads
__global__ void kan_combined(const float* __restrict__ x, const float* __restrict__ E,
                             const float* __restrict__ mask, const float* __restrict__ omiga,
                             float* __restrict__ comb, int in_n, int out_n, int F)
{
    long blk = blockIdx.x;
    int s = (int)(blk % SEQ);
    int o = (int)((blk / SEQ) % out_n);
    int b = (int)(blk / ((long)SEQ * out_n));
    int f = threadIdx.x;
    float acc = 0.f;
    for (int i = 0; i < in_n; ++i) {
        float coef = (E[(((long)b * in_n + i) * out_n + o) * SEQ + s]
                      + fabsf(omiga[(long)i * out_n + o]))
                   * mask[((long)b * in_n + i) * out_n + o];
        acc += coef * x[(((long)b * in_n + i) * SEQ + s) * F + f];
    }
    comb[(((long)b * out_n + o) * SEQ + s) * F + f] = acc / (float)in_n;
}

// LayerNorm over last dim D (optional affine indexed by o); in-place safe.
// grid = rows (B*out*SEQ), block 256
__global__ void ln_kernel(const float* __restrict__ in, float* __restrict__ out,
                          const float* __restrict__ w, const float* __restrict__ b,
                          int D, int out_n)
{
    long r = blockIdx.x;
    int o = (int)((r / SEQ) % out_n);
    const float* row = in + r * (long)D;
    float* orow = out + r * (long)D;
    __shared__ float sm[256];
    int tid = threadIdx.x;
    float s1 = 0.f;
    for (int j = tid; j < D; j += 256) s1 += row[j];
    sm[tid] = s1; __syncthreads();
    for (int st = 128; st; st >>= 1) { if (tid < st) sm[tid] += sm[tid + st]; __syncthreads(); }
    float mean = sm[0] / (float)D; __syncthreads();
    float s2 = 0.f;
    for (int j = tid; j < D; j += 256) { float d = row[j] - mean; s2 += d * d; }
    sm[tid] = s2; __syncthreads();
    for (int st = 128; st; st >>= 1) { if (tid < st) sm[tid] += sm[tid + st]; __syncthreads(); }
    float inv = rsqrtf(sm[0] / (float)D + 1e-5f);
    __syncthreads();
    for (int j = tid; j < D; j += 256) {
        float v = (row[j] - mean) * inv;
        if (w) v = v * w[(long)o * D + j] + b[(long)o * D + j];
        orow[j] = v;
    }
}

// K[b,o,s,i*P+p] = proj[b,i,s,p]*g[b,i,o] ; Q uses i+mid.  1 block per (b,o,s)
__global__ void build_kq(const float* __restrict__ proj, const float* __restrict__ g,
                         float* __restrict__ K, float* __restrict__ Q,
                         int in_n, int out_n, int P, int mid)
{
    long rb = blockIdx.x;
    int s = (int)(rb % SEQ);
    int o = (int)((rb / SEQ) % out_n);
    int b = (int)(rb / ((long)SEQ * out_n));
    int D = mid * P;
    for (int d = threadIdx.x; d < D; d += blockDim.x) {
        int i = d / P, p = d % P;
        float pk = proj[(((long)b * in_n + i) * SEQ + s) * P + p];
        float pq = proj[(((long)b * in_n + i + mid) * SEQ + s) * P + p];
        K[rb * (long)D + d] = pk * g[((long)b * in_n + i) * out_n + o];
        Q[rb * (long)D + d] = pq * g[((long)b * in_n + i + mid) * out_n + o];
    }
}

// softmax over last dim (256) with scale 1/(sqrt(d)*t), in-place; 1 block per row
__global__ void softmax_kernel(float* __restrict__ raw, const float* __restrict__ temp, float sqrtd)
{
    long r = blockIdx.x;
    float* row = raw + r * (long)SEQ;
    int tid = threadIdx.x;
    float t = fabsf(temp[0]) + 1e-4f;
    float scale = 1.0f / (sqrtd * t);
    __shared__ float sm[256];
    float v = row[tid] * scale;
    sm[tid] = v; __syncthreads();
    for (int st = 128; st; st >>= 1) { if (tid < st) sm[tid] = fmaxf(sm[tid], sm[tid + st]); __syncthreads(); }
    float mx = sm[0]; __syncthreads();
    float e = __expf(v - mx);
    sm[tid] = e; __syncthreads();
    for (int st = 128; st; st >>= 1) { if (tid < st) sm[tid] += sm[tid + st]; __syncthreads(); }
    row[tid] = e / sm[0];
}

// depthwise conv over xs[b, o*oh+d, 16,16] where xs[..,h,w] = xp[b,o,h*16+w,d]
// grid (B, out*oh), block 256
__global__ void dwconv(const float* __restrict__ xp, const float* __restrict__ w,
                       float* __restrict__ w3, int out_n, int oh, int k)
{
    int b = blockIdx.x, ch = blockIdx.y;
    int o = ch / oh, d = ch % oh;
    int t = threadIdx.x;
    int h = t >> 4, wq = t & 15;
    int pad = k / 2;
    float acc = 0.f;
    for (int dy = 0; dy < k; ++dy) {
        int hy = h + dy - pad;
        if (hy < 0 || hy > 15) continue;
        for (int dx = 0; dx < k; ++dx) {
            int wx = wq + dx - pad;
            if (wx < 0 || wx > 15) continue;
            acc += w[((long)ch * k + dy) * k + dx]
                 * xp[(((long)b * out_n + o) * SEQ + hy * 16 + wx) * oh + d];
        }
    }
    w3[(((long)b * out_n + o) * SEQ + h * 16 + wq) * oh + d] = acc;
}

// x_next = (attn? |beta|*W1xp + |alpha|*xp : xp) + |theta|*w3 + gamma*res
__global__ void combine_kernel(const float* __restrict__ xp, const float* __restrict__ w1xp,
                               const float* __restrict__ w3, const float* __restrict__ res,
                               const float* __restrict__ alpha, const float* __restrict__ beta,
                               const float* __restrict__ gamma, const float* __restrict__ theta,
                               float* __restrict__ xout, int out_n, int oh, int hasAttn, long total)
{
    long e = blockIdx.x * 256L + threadIdx.x;
    if (e >= total) return;
    int o = (int)((e / ((long)SEQ * oh)) % out_n);
    float base = hasAttn ? fabsf(beta[o]) * w1xp[e] + fabsf(alpha[o]) * xp[e] : xp[e];
    xout[e] = base + fabsf(theta[o]) * w3[e] + gamma[o] * res[e];
}

// out[b*10+o] = mean over S*oh ; grid = B*10, block 256
__global__ void final_mean(const float* __restrict__ x, float* __restrict__ out, int oh)
{
    int r = blockIdx.x;
    const float* base = x + (long)r * SEQ * oh;
    __shared__ float sm[256];
    int tid = threadIdx.x;
    float s = 0.f;
    int tot = SEQ * oh;
    for (int j = tid; j < tot; j += 256) s += base[j];
    sm[tid] = s; __syncthreads();
    for (int st = 128; st; st >>= 1) { if (tid < st) sm[tid] += sm[tid + st]; __syncthreads(); }
    if (tid == 0) out[r] = sm[0] / (float)tot;
}

// ---------------------------------------------------------------------------
extern "C" void kernel_launch(void* const* d_in, const int* in_sizes, int n_in,
                              void* d_out, int out_size, void* d_ws, size_t ws_size,
                              hipStream_t stream)
{
    (void)in_sizes; (void)n_in; (void)out_size; (void)ws_size;

    // ---- unpack inputs in setup_inputs() dict order ----
    int idx = 0;
    auto nxt = [&]() { return (const float*)d_in[idx++]; };
    struct BN { const float *g, *b, *m, *v; };
    auto getbn = [&]() { BN r; r.g = nxt(); r.b = nxt(); r.m = nxt(); r.v = nxt(); return r; };

    const float* X = nxt();
    const float* sc_w = nxt(); const float* sc_b = nxt(); BN sc_bn = getbn();
    const float* b1_w = nxt(); const float* b1_b = nxt(); BN b1_bn = getbn();
    const float* b2_w = nxt(); const float* b2_b = nxt(); BN b2_bn = getbn();
    const float* b3_w = nxt(); const float* b3_b = nxt(); BN b3_bn = getbn();
    const float* pc_w = nxt(); const float* pc_b = nxt(); BN pc_bn = getbn();

    static const int CFG[7][5] = {{1,4,128,64,7},{4,8,64,64,5},{8,16,64,128,5},
                                  {16,16,128,128,3},{16,20,128,128,3},{20,16,128,16,3},
                                  {16,10,16,1,1}};
    struct Layer {
        const float *sw,*bs,*tau,*temp,*omiga,*W2,*bias,*lnw,*lnb,*alpha,*beta,*gamma,*theta,*conv,*resW,*resb;
        int in, out, hid, oh, k; bool hasRes;
    } L[7];
    for (int l = 0; l < 7; ++l) {
        Layer& p = L[l];
        p.in = CFG[l][0]; p.out = CFG[l][1]; p.hid = CFG[l][2]; p.oh = CFG[l][3]; p.k = CFG[l][4];
        p.sw = nxt(); p.bs = nxt(); p.tau = nxt(); p.temp = nxt(); p.omiga = nxt();
        p.W2 = nxt(); p.bias = nxt(); p.lnw = nxt(); p.lnb = nxt();
        p.alpha = nxt(); p.beta = nxt(); p.gamma = nxt(); p.theta = nxt(); p.conv = nxt();
        p.hasRes = (p.hid != p.oh);
        if (p.hasRes) { p.resW = nxt(); p.resb = nxt(); } else { p.resW = nullptr; p.resb = nullptr; }
    }

    // ---- carve workspace ----
    char* base = (char*)d_ws;
    size_t off = 0;
    auto alloc = [&](long nfloat) -> float* {
        size_t a = (off + 255) & ~(size_t)255;
        off = a + (size_t)nfloat * sizeof(float);
        return (float*)(base + a);
    };
    const long XMAX  = 8L * 20 * SEQ * 128;   // 5,242,880
    const long KQMAX = 8L * SEQ * 20480;      // max out*d = 20480 -> 41,943,040
    const long RAWMX = 8L * 20 * SEQ * SEQ;   // 10,485,760
    const long WTMAX = 20L * 128 * 128;       // 327,680 (max out*hid*oh)
    float* xA    = alloc(XMAX);
    float* xB    = alloc(XMAX);
    float* projA = alloc(XMAX);
    float* projB = alloc(XMAX);
    float* comb  = alloc(XMAX);
    float* xlnB  = alloc(XMAX);
    float* w1xp  = alloc(XMAX);
    float* w3B   = alloc(XMAX);
    float* resB  = alloc(XMAX);
    float* xpT   = alloc(XMAX);               // xp transposed per (b,o): [oh,SEQ]
    float* Kb    = alloc(KQMAX);
    float* Qb    = alloc(KQMAX);
    float* rawB  = alloc(RAWMX);
    float* W2t   = alloc(WTMAX);              // W2 transposed per o: [oh,hid]
    float* resWt = alloc(WTMAX);              // resW transposed per o: [oh,hid]
    float* xnB   = alloc(8L * 20 * SEQ);
    float* aB    = alloc(8L * 20 * SEQ);
    float* Eb    = alloc(8L * 400 * SEQ);
    float* maskB = alloc(8L * 400);
    float* gB    = alloc(8L * 400);
    float* catB  = alloc(8L * 32 * 1024);
    float* scB   = alloc(8L * 32 * 1024);
    float* pcB   = alloc(8L * 32 * 1024);

    // ---- prep ----
    conv_bn_act<<<dim3(NB, 32), 256, 0, stream>>>(X, sc_w, sc_b, sc_bn.g, sc_bn.b, sc_bn.m, sc_bn.v,
                                                  scB, 5, 1, 0, 0, 32, 0);
    conv_bn_act<<<dim3(NB, 8),  256, 0, stream>>>(X, b1_w, b1_b, b1_bn.g, b1_bn.b, b1_bn.m, b1_bn.v,
                                                  catB, 5, 3, 1, 0, 32, 1);
    conv_bn_act<<<dim3(NB, 16), 256, 0, stream>>>(X, b2_w, b2_b, b2_bn.g, b2_bn.b, b2_bn.m, b2_bn.v,
                                                  catB, 5, 5, 2, 8, 32, 1);
    conv_bn_act<<<dim3(NB, 8),  256, 0, stream>>>(X, b3_w, b3_b, b3_bn.g, b3_bn.b, b3_bn.m, b3_bn.v,
                                                  catB, 5, 7, 3, 24, 32, 1);
    conv_bn_act<<<dim3(NB, 32), 256, 0, stream>>>(catB, pc_w, pc_b, pc_bn.g, pc_bn.b, pc_bn.m, pc_bn.v,
                                                  pcB, 32, 3, 1, 0, 32, 0);
    prep_finish<<<dim3(NB, 32), 256, 0, stream>>>(pcB, scB, xA);

    // ---- FastKAN layers ----
    float* x_cur = xA; float* x_nxt = xB;
    float* proj_prev = projA; float* proj_cur = projB;

    for (int l = 0; l < 7; ++l) {
        const Layer& p = L[l];
        const int in = p.in, out = p.out, hid = p.hid, oh = p.oh;
        const long rows = (long)NB * out * SEQ;
        const int ntiles = (oh + 15) / 16;
        const bool fulln = (oh % 16 == 0);

        kan_stats<<<dim3(NB, in), 256, 0, stream>>>(x_cur, xnB, aB, in, hid);
        kan_edge<<<dim3(NB, in, out), 256, 0, stream>>>(xnB, p.sw, p.bs, Eb, in, out);
        kan_ee<<<dim3(NB, in, out), 256, 0, stream>>>(Eb, aB, p.omiga, p.tau, p.temp,
                                                      maskB, gB, in, out);
        kan_combined<<<(unsigned)rows, hid, 0, stream>>>(x_cur, Eb, maskB, p.omiga, comb, in, out, hid);
        ln_kernel<<<(unsigned)rows, 256, 0, stream>>>(comb, xlnB, p.lnw, p.lnb, hid, out);

        // xp = xln @ W2 + bias   -> proj_cur  (W2 [hid,oh] pre-transposed to [oh,hid])
        transpose_kernel<<<dim3(out, hid), 256, 0, stream>>>(p.W2, W2t, hid, oh, (long)hid * oh);
        if (fulln)
            wmma_gemm_nt<1><<<dim3(16, ntiles, NB * out), 32, 0, stream>>>(
                xlnB, W2t, p.bias, proj_cur, SEQ, oh, hid,
                (long)SEQ * hid, (long)hid * oh, (long)SEQ * oh, out, 1);
        else
            wmma_gemm_nt<0><<<dim3(16, ntiles, NB * out), 32, 0, stream>>>(
                xlnB, W2t, p.bias, proj_cur, SEQ, oh, hid,
                (long)SEQ * hid, (long)hid * oh, (long)SEQ * oh, out, 1);

        int hasAttn = (l > 0);
        if (hasAttn) {
            int P = CFG[l - 1][3];       // oh of previous layer
            int mid = in / 2;
            int d = mid * P;
            build_kq<<<(unsigned)rows, 256, 0, stream>>>(proj_prev, gB, Kb, Qb, in, out, P, mid);
            ln_kernel<<<(unsigned)rows, 256, 0, stream>>>(Kb, Kb, nullptr, nullptr, d, out);
            ln_kernel<<<(unsigned)rows, 256, 0, stream>>>(Qb, Qb, nullptr, nullptr, d, out);
            // raw[s,t] = sum_d K[s,d]*Q[t,d]  -> Q already in [N,K] layout
            wmma_gemm_nt<1><<<dim3(16, 16, NB * out), 32, 0, stream>>>(
                Kb, Qb, nullptr, rawB, SEQ, SEQ, d,
                (long)SEQ * d, (long)SEQ * d, (long)SEQ * SEQ, 0, 0);
            softmax_kernel<<<(unsigned)rows, 256, 0, stream>>>(rawB, p.temp, sqrtf((float)d));
            // W1 @ xp : pre-transpose xp [SEQ,oh] -> [oh,SEQ] per (b,o)
            transpose_kernel<<<dim3(NB * out, SEQ), 256, 0, stream>>>(
                proj_cur, xpT, SEQ, oh, (long)SEQ * oh);
            if (fulln)
                wmma_gemm_nt<1><<<dim3(16, ntiles, NB * out), 32, 0, stream>>>(
                    rawB, xpT, nullptr, w1xp, SEQ, oh, SEQ,
                    (long)SEQ * SEQ, (long)SEQ * oh, (long)SEQ * oh, 0, 0);
            else
                wmma_gemm_nt<0><<<dim3(16, ntiles, NB * out), 32, 0, stream>>>(
                    rawB, xpT, nullptr, w1xp, SEQ, oh, SEQ,
                    (long)SEQ * SEQ, (long)SEQ * oh, (long)SEQ * oh, 0, 0);
        }

        dwconv<<<dim3(NB, out * oh), 256, 0, stream>>>(proj_cur, p.conv, w3B, out, oh, p.k);

        const float* resPtr = comb;
        if (p.hasRes) {
            transpose_kernel<<<dim3(out, hid), 256, 0, stream>>>(p.resW, resWt, hid, oh, (long)hid * oh);
            if (fulln)
                wmma_gemm_nt<1><<<dim3(16, ntiles, NB * out), 32, 0, stream>>>(
                    comb, resWt, p.resb, resB, SEQ, oh, hid,
                    (long)SEQ * hid, (long)hid * oh, (long)SEQ * oh, out, 2);
            else
                wmma_gemm_nt<0><<<dim3(16, ntiles, NB * out), 32, 0, stream>>>(
                    comb, resWt, p.resb, resB, SEQ, oh, hid,
                    (long)SEQ * hid, (long)hid * oh, (long)SEQ * oh, out, 2);
            resPtr = resB;
        }

        long total = rows * oh;
        combine_kernel<<<(unsigned)((total + 255) / 256), 256, 0, stream>>>(
            proj_cur, w1xp, w3B, resPtr, p.alpha, p.beta, p.gamma, p.theta,
            x_nxt, out, oh, hasAttn, total);

        { float* t = x_cur; x_cur = x_nxt; x_nxt = t; }
        { float* t = proj_prev; proj_prev = proj_cur; proj_cur = t; }
    }

    // final mean over (S, oh=1) -> [8,10]
    final_mean<<<NB * 10, 256, 0, stream>>>(x_cur, (float*)d_out, 1);
}